// GIN_73521250173172
// MI455X (gfx1250) — compile-verified
//
#include <hip/hip_runtime.h>
#include <hip/hip_bf16.h>
#include <cstdint>

#define D_HID 128
#define D_OUT 64
#define BN_EPS 1e-5f

typedef float v2f __attribute__((ext_vector_type(2)));
typedef float v8f __attribute__((ext_vector_type(8)));

// ---------------------------------------------------------------- copy (agg = x)
__global__ void copy_f4(const float4* __restrict__ src, float4* __restrict__ dst, int n4) {
    int i = blockIdx.x * blockDim.x + threadIdx.x;
    if (i < n4) dst[i] = src[i];
}

// ---------------------------------------------------------------- edge scatter-add
// one wave32 per edge: 32 lanes x float4 = 128 features
__global__ void scatter_add_edges(const float* __restrict__ x, float* __restrict__ agg,
                                  const long long* __restrict__ esrc,
                                  const long long* __restrict__ edst, int E) {
    int gid  = blockIdx.x * blockDim.x + threadIdx.x;
    int e    = gid >> 5;
    int lane = gid & 31;
    if (e >= E) return;
    int s = (int)esrc[e];
    int t = (int)edst[e];
    const float4 v = *(const float4*)(x + (size_t)s * D_HID + lane * 4);
    float* p = agg + (size_t)t * D_HID + lane * 4;
    unsafeAtomicAdd(p + 0, v.x);
    unsafeAtomicAdd(p + 1, v.y);
    unsafeAtomicAdd(p + 2, v.z);
    unsafeAtomicAdd(p + 3, v.w);
}

// ---------------------------------------------------------------- fp32 WMMA GEMM
// C[N x Kout] = A[N x 128] @ W[128 x Kout] + bias
// grid.x = N/16 row tiles; block = (Kout/16) waves, each wave one 16x16 tile.
__global__ void gemm_wmma_f32(const float* __restrict__ A, const float* __restrict__ W,
                              const float* __restrict__ bias, float* __restrict__ C,
                              int Kout) {
    const int K   = 128;
    int wave = threadIdx.x >> 5;
    int lane = threadIdx.x & 31;
    int half = lane >> 4;    // 0 -> K pair {k,k+1}; 1 -> {k+2,k+3}   (ISA 7.12.2)
    int lm   = lane & 15;
    int m0   = blockIdx.x * 16;
    int n0   = wave * 16;

    const float* arow = A + (size_t)(m0 + lm) * K + 2 * half;       // A[m][k+2*half]
    const float* wcol = W + (size_t)(2 * half) * Kout + n0 + lm;    // W[k+2*half][n]

    v8f acc = {};
#pragma unroll
    for (int k = 0; k < K; k += 4) {
        v2f a, b;
        a.x = arow[k + 0];
        a.y = arow[k + 1];
        b.x = wcol[(size_t)(k + 0) * Kout];
        b.y = wcol[(size_t)(k + 1) * Kout];
        // v_wmma_f32_16x16x4_f32: (neg_a, A, neg_b, B, c_mod, C, reuse_a, reuse_b)
        acc = __builtin_amdgcn_wmma_f32_16x16x4_f32(false, a, false, b,
                                                    (short)0, acc, false, false);
    }

    float bv = bias[n0 + lm];
#pragma unroll
    for (int r = 0; r < 8; ++r) {
        int row = m0 + r + 8 * half;    // VGPR r: lanes 0-15 -> M=r, lanes 16-31 -> M=8+r
        C[(size_t)row * Kout + n0 + lm] = acc[r] + bv;
    }
}

// ---------------------------------------------------------------- BN column stats
__global__ void bn_stats(const float* __restrict__ h, float* __restrict__ mean,
                         float* __restrict__ rstd, int Nn) {
    __shared__ float sh[256], sh2[256];
    int c = blockIdx.x;
    float s = 0.f, s2 = 0.f;
    for (int r = threadIdx.x; r < Nn; r += blockDim.x) {
        float v = h[(size_t)r * D_HID + c];
        s += v; s2 += v * v;
    }
    sh[threadIdx.x] = s; sh2[threadIdx.x] = s2;
    __syncthreads();
    for (int off = 128; off > 0; off >>= 1) {
        if (threadIdx.x < off) {
            sh[threadIdx.x]  += sh[threadIdx.x + off];
            sh2[threadIdx.x] += sh2[threadIdx.x + off];
        }
        __syncthreads();
    }
    if (threadIdx.x == 0) {
        float m   = sh[0] / (float)Nn;
        float var = sh2[0] / (float)Nn - m * m;
        mean[c] = m;
        rstd[c] = rsqrtf(var + BN_EPS);
    }
}

// ---------------------------------------------------------------- BN apply + ReLU
__global__ void bn_apply_relu(float* __restrict__ h, const float* __restrict__ mean,
                              const float* __restrict__ rstd, const float* __restrict__ g,
                              const float* __restrict__ be, int n4) {
    int i = blockIdx.x * blockDim.x + threadIdx.x;
    if (i >= n4) return;
    float4 v = ((float4*)h)[i];
    int c = (i & (D_HID / 4 - 1)) * 4;
    v.x = fmaxf(0.f, (v.x - mean[c + 0]) * rstd[c + 0] * g[c + 0] + be[c + 0]);
    v.y = fmaxf(0.f, (v.y - mean[c + 1]) * rstd[c + 1] * g[c + 1] + be[c + 1]);
    v.z = fmaxf(0.f, (v.z - mean[c + 2]) * rstd[c + 2] * g[c + 2] + be[c + 2]);
    v.w = fmaxf(0.f, (v.w - mean[c + 3]) * rstd[c + 3] * g[c + 3] + be[c + 3]);
    ((float4*)h)[i] = v;
}

// ---------------------------------------------------------------- row log-softmax (64 cols)
__global__ void log_softmax64(float* __restrict__ o, int Nn) {
    int gid  = blockIdx.x * blockDim.x + threadIdx.x;
    int r    = gid >> 5;
    int lane = gid & 31;
    if (r >= Nn) return;
    float* row = o + (size_t)r * D_OUT;
    float v0 = row[lane], v1 = row[lane + 32];
    float m = fmaxf(v0, v1);
#pragma unroll
    for (int off = 16; off > 0; off >>= 1) m = fmaxf(m, __shfl_xor(m, off, 32));
    float s = __expf(v0 - m) + __expf(v1 - m);
#pragma unroll
    for (int off = 16; off > 0; off >>= 1) s += __shfl_xor(s, off, 32);
    float lse = m + __logf(s);
    row[lane]      = v0 - lse;
    row[lane + 32] = v1 - lse;
}

// ================================================================ host launcher
extern "C" void kernel_launch(void* const* d_in, const int* in_sizes, int n_in,
                              void* d_out, int out_size, void* d_ws, size_t ws_size,
                              hipStream_t stream) {
    const float*     x   = (const float*)d_in[0];
    const long long* ei  = (const long long*)d_in[1];   // int64 (2, E)
    const float* W1 = (const float*)d_in[2];  const float* b1  = (const float*)d_in[3];
    const float* g1 = (const float*)d_in[4];  const float* be1 = (const float*)d_in[5];
    const float* W2 = (const float*)d_in[6];  const float* b2  = (const float*)d_in[7];
    const float* g2 = (const float*)d_in[8];  const float* be2 = (const float*)d_in[9];
    const float* W3 = (const float*)d_in[10]; const float* b3  = (const float*)d_in[11];

    int Nn = in_sizes[0] / D_HID;        // 50000
    int E  = in_sizes[1] / 2;            // 600000
    const long long* esrc = ei;
    const long long* edst = ei + E;

    float* agg  = (float*)d_ws;                       // N x 128
    float* feat = agg + (size_t)Nn * D_HID;           // N x 128
    float* mean = feat + (size_t)Nn * D_HID;          // 128
    float* rstd = mean + D_HID;                       // 128
    float* out  = (float*)d_out;                      // N x 64

    int n4       = Nn * D_HID / 4;
    int copyGrid = (n4 + 255) / 256;
    int scatGrid = (int)(((size_t)E * 32 + 255) / 256);
    int rowTiles = (Nn + 15) / 16;
    dim3 blk(256);

    // ---- layer 1: agg = x + sum_{j->i} x_j ; feat = agg@W1+b1 ; BN+ReLU
    copy_f4<<<copyGrid, blk, 0, stream>>>((const float4*)x, (float4*)agg, n4);
    scatter_add_edges<<<scatGrid, blk, 0, stream>>>(x, agg, esrc, edst, E);
    gemm_wmma_f32<<<rowTiles, dim3((D_HID / 16) * 32), 0, stream>>>(agg, W1, b1, feat, D_HID);
    bn_stats<<<D_HID, 256, 0, stream>>>(feat, mean, rstd, Nn);
    bn_apply_relu<<<copyGrid, blk, 0, stream>>>(feat, mean, rstd, g1, be1, n4);

    // ---- layer 2
    copy_f4<<<copyGrid, blk, 0, stream>>>((const float4*)feat, (float4*)agg, n4);
    scatter_add_edges<<<scatGrid, blk, 0, stream>>>(feat, agg, esrc, edst, E);
    gemm_wmma_f32<<<rowTiles, dim3((D_HID / 16) * 32), 0, stream>>>(agg, W2, b2, feat, D_HID);
    bn_stats<<<D_HID, 256, 0, stream>>>(feat, mean, rstd, Nn);
    bn_apply_relu<<<copyGrid, blk, 0, stream>>>(feat, mean, rstd, g2, be2, n4);

    // ---- layer 3: straight to d_out (N x 64), then log-softmax in place
    copy_f4<<<copyGrid, blk, 0, stream>>>((const float4*)feat, (float4*)agg, n4);
    scatter_add_edges<<<scatGrid, blk, 0, stream>>>(feat, agg, esrc, edst, E);
    gemm_wmma_f32<<<rowTiles, dim3((D_OUT / 16) * 32), 0, stream>>>(agg, W3, b3, out, D_OUT);
    log_softmax64<<<(int)(((size_t)Nn * 32 + 255) / 256), blk, 0, stream>>>(out, Nn);
}